// SelectedFrameReconstructor_16389595201720
// MI455X (gfx1250) — compile-verified
//
#include <hip/hip_runtime.h>

// ---------------------------------------------------------------------------
// SelectedFrameReconstructor for MI455X (gfx1250, wave32).
// Projection uses V_WMMA_F32_16X16X4_F32 (A: 16 pts x 4 homog, B: 4 x 16
// view-component columns). Depth z-test = monotone-keyed u32 atomic min.
// Splat = f32 atomic adds. All L2-resident; atomic-bound workload.
// ---------------------------------------------------------------------------

namespace {
constexpr int   kKimg  = 2;
constexpr int   kH     = 128;
constexpr int   kW     = 128;
constexpr int   kS     = 64;
constexpr int   kHW    = kH * kW;          // 16384
constexpr int   kN     = kKimg * kHW;      // 32768 points
constexpr float kEPS   = 1e-7f;
constexpr float kConfTh = 0.5f;
constexpr unsigned kInfKey = 0xFF800000u;  // key(+inf)

// workspace layout, in floats
constexpr int kOffM    = 0;                    // 64*12 = 768 used, 1024 reserved
constexpr int kOffDep  = 1024;                 // S*HW u32 depth keys
constexpr int kOffRgb  = kOffDep + kS * kHW;   // 3*S*HW f32
constexpr int kOffConf = kOffRgb + 3 * kS * kHW; // S*HW f32
}

typedef __attribute__((ext_vector_type(2))) float v2f;
typedef __attribute__((ext_vector_type(8))) float v8f;

// Monotone float -> u32 key (total order matches float order for non-NaN).
__device__ __forceinline__ unsigned fkey(float f) {
  unsigned b = __float_as_uint(f);
  return (b & 0x80000000u) ? ~b : (b | 0x80000000u);
}
__device__ __forceinline__ float fdec(unsigned k) {
  return __uint_as_float((k & 0x80000000u) ? (k ^ 0x80000000u) : ~k);
}

// ---- fuse M[s] = intrinsic[s] (3x3) @ extrinsic[s] (3x4). Row 2 stays exact
// ([0,0,1,t_z]) because K row2 = [0,0,1], so z == uvh.z bitwise. ----
__global__ void k_prep(const float* __restrict__ extr,
                       const float* __restrict__ intr,
                       float* __restrict__ Mmat) {
  const int s = threadIdx.x;
  if (s >= kS) return;
  const float* E  = extr + s * 12;  // 3x4
  const float* Kc = intr + s * 9;   // 3x3
#pragma unroll
  for (int i = 0; i < 3; ++i) {
#pragma unroll
    for (int j = 0; j < 4; ++j) {
      float acc = Kc[i * 3 + 0] * E[0 * 4 + j] +
                  Kc[i * 3 + 1] * E[1 * 4 + j] +
                  Kc[i * 3 + 2] * E[2 * 4 + j];
      Mmat[s * 12 + i * 4 + j] = acc;
    }
  }
}

__global__ void k_init(unsigned* __restrict__ depth,
                       float* __restrict__ rgbacc,
                       float* __restrict__ confacc) {
  const int i = blockIdx.x * blockDim.x + threadIdx.x;
  if (i >= kS * kHW) return;
  depth[i]          = kInfKey;
  confacc[i]        = 0.f;
  rgbacc[i * 3 + 0] = 0.f;
  rgbacc[i * 3 + 1] = 0.f;
  rgbacc[i * 3 + 2] = 0.f;
}

// ---- passes. PHASE 0: unconditional depth atomic-min over ALL pairs
//      (reference mins over invalid entries too). PHASE 1: bilinear splat for
//      valid pairs whose z equals the depth-buffer minimum. ----
template <int PHASE>
__global__ void __launch_bounds__(64)
k_pass(const float* __restrict__ pts,      // (N,3)
       const float* __restrict__ conf,     // (N)
       const float* __restrict__ images,   // (K,3,H,W)
       const float* __restrict__ Mmat,     // (S,3,4)
       unsigned* __restrict__ depth,
       float* __restrict__ rgbacc,
       float* __restrict__ confacc) {
  __shared__ float lds[2][16 * 256];       // per-wave: 16 pts x 64 views x 4
  const int lane = threadIdx.x & 31;
  const int wid  = threadIdx.x >> 5;
  const int half = lane >> 4;              // 0: K=0/1, 1: K=2/3
  const int l    = lane & 15;
  const int p0   = blockIdx.x * 32 + wid * 16;
  float* luvz = lds[wid];

  // A operand (16x4 f32): lane half selects K pair per ISA layout.
  const int   p  = p0 + l;
  const float px = pts[p * 3 + 0];
  const float py = pts[p * 3 + 1];
  const float pz = pts[p * 3 + 2];
  v2f a;
  a.x = half ? pz : px;
  a.y = half ? 1.0f : py;
  const int k0 = half ? 2 : 0;

  // 16 tiles of 16 columns cover 64 views x 4 (u,v,z,pad) components.
  for (int t = 0; t < 16; ++t) {
    const int g    = t * 16 + l;           // global column = B column = C col
    const int s    = g >> 2;
    const int comp = g & 3;
    const float* Ms = Mmat + s * 12 + (comp < 3 ? comp : 2) * 4;
    const float b0 = Ms[k0];
    const float b1 = Ms[k0 + 1];
    v2f b;
    b.x = (comp < 3) ? b0 : 0.f;
    b.y = (comp < 3) ? b1 : 0.f;
    v8f c = {};
    c = __builtin_amdgcn_wmma_f32_16x16x4_f32(
        /*neg_a=*/false, a, /*neg_b=*/false, b,
        /*c_mod=*/(short)0, c, /*reuse_a=*/false, /*reuse_b=*/false);
#pragma unroll
    for (int r = 0; r < 8; ++r)            // C row r (+8 for lane half)
      luvz[(r + 8 * half) * 256 + g] = c[r];
  }
  __syncthreads();

  // consume 16 pts x 64 views = 1024 pairs with 32 lanes
  for (int it = 0; it < 32; ++it) {
    const int q  = it * 32 + lane;
    const int pl = q >> 6;                 // local point
    const int s  = q & 63;                 // view
    const float uh = luvz[pl * 256 + s * 4 + 0];
    const float vh = luvz[pl * 256 + s * 4 + 1];
    const float z  = luvz[pl * 256 + s * 4 + 2];
    const float den = z + kEPS;
    const float u = uh / den;
    const float v = vh / den;
    const float uc = fminf(fmaxf(u, 0.f), (float)(kW - 1));
    const float vc = fminf(fmaxf(v, 0.f), (float)(kH - 1));
    const int u0 = (int)uc;
    const int v0 = (int)vc;
    const int pix = s * kHW + v0 * kW + u0;

    if (PHASE == 0) {
      atomicMin(&depth[pix], fkey(z));
    } else {
      const int   pg = p0 + pl;
      const float cf = conf[pg];
      const bool valid = (z > 0.f) && (u >= 0.f) && (u < (float)kW) &&
                         (v >= 0.f) && (v < (float)kH) && (cf >= kConfTh);
      if (valid && depth[pix] == fkey(z)) {
        const int u1 = min(u0 + 1, kW - 1);
        const int v1 = min(v0 + 1, kH - 1);
        const float du = uc - (float)u0;
        const float dv = vc - (float)v0;
        const float w00 = (1.f - du) * (1.f - dv);
        const float w01 = (1.f - du) * dv;
        const float w10 = du * (1.f - dv);
        const float w11 = du * dv;
        const int k  = pg / kHW;
        const int hw = pg - k * kHW;
        const float r0  = images[(k * 3 + 0) * kHW + hw];
        const float g0  = images[(k * 3 + 1) * kHW + hw];
        const float b0c = images[(k * 3 + 2) * kHW + hw];
        const int base = s * kHW;
        const float ws4[4] = {w00, w01, w10, w11};
        const int   px4[4] = {base + v0 * kW + u0, base + v1 * kW + u0,
                              base + v0 * kW + u1, base + v1 * kW + u1};
#pragma unroll
        for (int i = 0; i < 4; ++i) {
          const float wgt = ws4[i];
          const int   pp  = px4[i];
          atomicAdd(&rgbacc[pp * 3 + 0], wgt * r0);
          atomicAdd(&rgbacc[pp * 3 + 1], wgt * g0);
          atomicAdd(&rgbacc[pp * 3 + 2], wgt * b0c);
          atomicAdd(&confacc[pp], wgt * cf);
        }
      }
    }
  }
}

__global__ void k_final(const unsigned* __restrict__ depth,
                        const float* __restrict__ rgbacc,
                        const float* __restrict__ confacc,
                        float* __restrict__ out) {
  const int i = blockIdx.x * blockDim.x + threadIdx.x;
  if (i >= kS * kHW) return;
  const int s  = i / kHW;
  const int hw = i - s * kHW;
  float r = rgbacc[i * 3 + 0];
  float g = rgbacc[i * 3 + 1];
  float b = rgbacc[i * 3 + 2];
  const bool over = (r > 1.f) || (g > 1.f) || (b > 1.f);
  r = fminf(fmaxf(r, 0.f), 1.f);
  g = fminf(fmaxf(g, 0.f), 1.f);
  b = fminf(fmaxf(b, 0.f), 1.f);
  out[(s * 3 + 0) * kHW + hw] = r;   // rgb_map (S,3,H,W)
  out[(s * 3 + 1) * kHW + hw] = g;
  out[(s * 3 + 2) * kHW + hw] = b;
  const float d = fdec(depth[i]);
  out[3 * kS * kHW + i] = d;                                 // depth_map
  out[4 * kS * kHW + i] = confacc[i];                        // conf_map
  out[5 * kS * kHW + i] = (d < INFINITY && !over) ? 1.f : 0.f; // mask_map
}

extern "C" void kernel_launch(void* const* d_in, const int* in_sizes, int n_in,
                              void* d_out, int out_size, void* d_ws,
                              size_t ws_size, hipStream_t stream) {
  (void)in_sizes; (void)n_in; (void)out_size; (void)ws_size;
  const float* images = (const float*)d_in[0];  // (K,3,H,W)
  const float* pts    = (const float*)d_in[1];  // (K,H,W,3)
  const float* conf   = (const float*)d_in[2];  // (K,H,W)
  const float* extr   = (const float*)d_in[3];  // (S,3,4)
  const float* intr   = (const float*)d_in[4];  // (S,3,3)
  float* out = (float*)d_out;
  float* ws  = (float*)d_ws;

  float*    Mmat    = ws + kOffM;
  unsigned* depth   = (unsigned*)(ws + kOffDep);
  float*    rgbacc  = ws + kOffRgb;
  float*    confacc = ws + kOffConf;

  k_prep<<<1, 64, 0, stream>>>(extr, intr, Mmat);
  k_init<<<(kS * kHW + 255) / 256, 256, 0, stream>>>(depth, rgbacc, confacc);
  k_pass<0><<<kN / 32, 64, 0, stream>>>(pts, conf, images, Mmat, depth, rgbacc,
                                        confacc);
  k_pass<1><<<kN / 32, 64, 0, stream>>>(pts, conf, images, Mmat, depth, rgbacc,
                                        confacc);
  k_final<<<(kS * kHW + 255) / 256, 256, 0, stream>>>(depth, rgbacc, confacc,
                                                      out);
}